// EnergyMetric_10204842295392
// MI455X (gfx1250) — compile-verified
//
#include <hip/hip_runtime.h>
#include <hip/hip_bf16.h>
#include <math.h>

// ---------------------------------------------------------------------------
// EnergyMetric: iterative weighted Procrustes (SVD replaced by Newton-Schulz
// polar iteration, same result T = U @ Vt) + pairwise energy loss.
// Heavy math on V_WMMA_F32_16X16X4_F32; the 20-step NS loop is fused into a
// single persistent-workgroup kernel (1 WG of 32 waves per batch) to kill
// launch latency, which dominates at batch=8 on MI455X.
// ---------------------------------------------------------------------------

typedef __attribute__((ext_vector_type(2))) float v2f;
typedef __attribute__((ext_vector_type(8))) float v8f;

#define B_   8
#define C_   64
#define R_   4
#define D_   256
#define NF_  1024            // C_ * R_ * R_
#define TD2_ 65536           // 256*256
#define NITER 100
#define NS_ITERS 20          // even -> final T lands back in T0

__device__ __forceinline__ v8f wmma4(v2f a, v2f b, v8f c) {
  // D(16x16,f32) = A(16x4,f32) x B(4x16,f32) + C
  return __builtin_amdgcn_wmma_f32_16x16x4_f32(
      /*neg_a=*/false, a, /*neg_b=*/false, b,
      /*c_mod=*/(short)0, c, /*reuse_a=*/false, /*reuse_b=*/false);
}

// -------------------- M = Xf^T diag(w^2) Yf  (per batch 256x1024x256) ------
__global__ void k_gemm_M(const float* __restrict__ X, const float* __restrict__ Y,
                         const float* __restrict__ w2, float* __restrict__ M) {
  const int wave = blockIdx.x * (blockDim.x >> 5) + (threadIdx.x >> 5);
  const int lane = threadIdx.x & 31;
  const int l16 = lane & 15, lhi = lane >> 4;
  const int b   = wave >> 8;           // 256 tiles per batch
  const int tile = wave & 255;
  const int tm = tile >> 4, tn = tile & 15;
  const int d0 = tm * 16 + l16;        // A column this lane reads (Xf^T row)
  const int e0 = tn * 16 + l16;        // B column this lane reads
  const float* Xb  = X  + b * (C_ * R_ * D_);
  const float* Yb  = Y  + b * (C_ * R_ * D_);
  const float* w2b = w2 + b * NF_;
  v8f acc = {};
  #pragma unroll 4
  for (int n0 = 0; n0 < NF_; n0 += 4) {
    v2f a, bv;
    #pragma unroll
    for (int v = 0; v < 2; ++v) {
      const int r  = n0 + v + 2 * lhi;           // flattened-product row
      const int c  = r >> 4, p = r & 15;
      const int rx = p >> 2, ry = p & 3;
      a[v]  = Xb[(c * R_ + rx) * D_ + d0];
      bv[v] = Yb[(c * R_ + ry) * D_ + e0] * w2b[r];
    }
    acc = wmma4(a, bv, acc);
  }
  float* Mb = M + b * TD2_;
  #pragma unroll
  for (int v = 0; v < 8; ++v)
    Mb[(tm * 16 + v + 8 * lhi) * D_ + tn * 16 + l16] = acc[v];
}

// -------------------- WMMA tile helpers for the fused NS kernel ------------
__device__ __forceinline__ void ata_tile(const float* Tb, float* Ab,
                                         int tile, int l16, int lhi) {
  const int ti = tile >> 4, tj = tile & 15;
  const int i0 = ti * 16 + l16, j0 = tj * 16 + l16;
  v8f acc = {};
  #pragma unroll 4
  for (int k0 = 0; k0 < D_; k0 += 4) {
    v2f a, bv;
    #pragma unroll
    for (int v = 0; v < 2; ++v) {
      const int kr = k0 + v + 2 * lhi;
      a[v]  = Tb[kr * D_ + i0];
      bv[v] = Tb[kr * D_ + j0];
    }
    acc = wmma4(a, bv, acc);
  }
  #pragma unroll
  for (int v = 0; v < 8; ++v)
    Ab[(ti * 16 + v + 8 * lhi) * D_ + tj * 16 + l16] = acc[v];
}

__device__ __forceinline__ void nsup_tile(const float* Tb, const float* Ab,
                                          float* Tn, int tile, int l16, int lhi) {
  const int ti = tile >> 4, tj = tile & 15;
  v8f acc = {};
  #pragma unroll 4
  for (int k0 = 0; k0 < D_; k0 += 4) {
    v2f a, bv;
    #pragma unroll
    for (int v = 0; v < 2; ++v) {
      const int kr = k0 + v + 2 * lhi;
      a[v]  = Tb[(ti * 16 + l16) * D_ + kr];
      bv[v] = Ab[kr * D_ + tj * 16 + l16];
    }
    acc = wmma4(a, bv, acc);
  }
  #pragma unroll
  for (int v = 0; v < 8; ++v) {
    const int row = ti * 16 + v + 8 * lhi, col = tj * 16 + l16;
    Tn[row * D_ + col] = 1.5f * Tb[row * D_ + col] - 0.5f * acc[v];
  }
}

// ---- fused: frobenius norm + scale + 20 Newton-Schulz steps, 1 WG/batch ---
// A may alias M (M is dead after the scale phase).
__global__ void __launch_bounds__(1024) k_ns_polar(const float* M,
                                                   float* __restrict__ T0,
                                                   float* __restrict__ T1,
                                                   float* A) {
  __shared__ float red[1024];
  const int b = blockIdx.x;
  const int tid = threadIdx.x;
  const float* Mb = M + b * TD2_;
  // Frobenius norm of M_b
  float s = 0.f;
  for (int i = tid; i < TD2_; i += 1024) { float m = Mb[i]; s += m * m; }
  red[tid] = s; __syncthreads();
  for (int st = 512; st > 0; st >>= 1) {
    if (tid < st) red[tid] += red[tid + st];
    __syncthreads();
  }
  const float inv = rsqrtf(fmaxf(red[0], 1e-30f));
  // T = M / ||M||_F  (sigma in (0,1] -> NS converges to polar factor U V^T)
  float* Tc = T0 + b * TD2_;
  float* Tn = T1 + b * TD2_;
  float* Ab = A + b * TD2_;
  for (int i = tid; i < TD2_; i += 1024) Tc[i] = Mb[i] * inv;
  __threadfence(); __syncthreads();

  const int wave = tid >> 5, lane = tid & 31;
  const int l16 = lane & 15, lhi = lane >> 4;
  for (int ns = 0; ns < NS_ITERS; ++ns) {
    // A = T^T T : 256 tiles over 32 waves
    #pragma unroll 1
    for (int t = 0; t < 8; ++t) ata_tile(Tc, Ab, wave * 8 + t, l16, lhi);
    __threadfence(); __syncthreads();
    // Tn = 1.5 T - 0.5 T A
    #pragma unroll 1
    for (int t = 0; t < 8; ++t) nsup_tile(Tc, Ab, Tn, wave * 8 + t, l16, lhi);
    __threadfence(); __syncthreads();
    float* tmp = Tc; Tc = Tn; Tn = tmp;
  }
  // NS_ITERS even -> result resides in T0
}

// ------------- rsum[b,n] += rowwise ||Xf - Yf*T||^2 (fused GEMM) -----------
__global__ void k_resid(const float* __restrict__ X, const float* __restrict__ Y,
                        const float* __restrict__ T, float* __restrict__ rsum) {
  const int wave = blockIdx.x * (blockDim.x >> 5) + (threadIdx.x >> 5);
  const int lane = threadIdx.x & 31;
  const int l16 = lane & 15, lhi = lane >> 4;
  const int b = wave >> 10;            // 1024 tiles per batch (64 x 16)
  const int tile = wave & 1023;
  const int tm = tile >> 4, tn = tile & 15;
  const int m0 = tm * 16;
  const int e0 = tn * 16 + l16;
  const float* Xb = X + b * (C_ * R_ * D_);
  const float* Yb = Y + b * (C_ * R_ * D_);
  const float* Tb = T + b * TD2_;
  const int n  = m0 + l16;
  const int c  = n >> 4, p = n & 15, ry = p & 3;
  const float* yrow = Yb + (c * R_ + ry) * D_;
  v8f acc = {};
  #pragma unroll 4
  for (int k0 = 0; k0 < D_; k0 += 4) {
    v2f a, bv;
    #pragma unroll
    for (int v = 0; v < 2; ++v) {
      const int kr = k0 + v + 2 * lhi;
      a[v]  = yrow[kr];
      bv[v] = Tb[kr * D_ + e0];
    }
    acc = wmma4(a, bv, acc);
  }
  #pragma unroll
  for (int v = 0; v < 8; ++v) {
    const int row = m0 + v + 8 * lhi;
    const int c2 = row >> 4, p2 = row & 15, rx = p2 >> 2;
    const float xv = Xb[(c2 * R_ + rx) * D_ + e0];
    const float dd = xv - acc[v];
    float sq = dd * dd;
    #pragma unroll
    for (int s = 8; s >= 1; s >>= 1) sq += __shfl_xor(sq, s, 16);
    if (l16 == 0) atomicAdd(&rsum[b * NF_ + row], sq);
  }
}

// -------------------- small utility kernels --------------------------------
__global__ void k_fill(float* p, float v, int n) {
  int i = blockIdx.x * blockDim.x + threadIdx.x;
  if (i < n) p[i] = v;
}

// w2 = 1/max(rsum^(1/4), tol)^2 ; also re-zeroes rsum for the next iteration
__global__ void k_w2up(float* __restrict__ rsum, float* __restrict__ w2) {
  const int i = blockIdx.x * blockDim.x + threadIdx.x;
  if (i >= B_ * NF_) return;
  const float r = rsum[i];
  rsum[i] = 0.f;
  const float q = sqrtf(sqrtf(fmaxf(r, 0.f)));
  const float m = fmaxf(q, 1e-6f);
  w2[i] = 1.f / (m * m);
}

// pairwise energy over repeat dim: 8*64*6 = 3072 pairs -> out[b] += |z1-z2|/384
__global__ void k_pair_energy(const float* __restrict__ Z, float* __restrict__ out) {
  const int wave = blockIdx.x * (blockDim.x >> 5) + (threadIdx.x >> 5);
  const int lane = threadIdx.x & 31;
  const int b = wave / 384, rem = wave % 384, c = rem / 6, pr = rem % 6;
  const int pi[6] = {0, 0, 0, 1, 1, 2};
  const int pj[6] = {1, 2, 3, 2, 3, 3};
  const float* z1 = Z + ((b * C_ + c) * R_ + pi[pr]) * D_;
  const float* z2 = Z + ((b * C_ + c) * R_ + pj[pr]) * D_;
  float s = 0.f;
  for (int t = lane; t < D_; t += 32) { float d = z1[t] - z2[t]; s += d * d; }
  #pragma unroll
  for (int st = 16; st >= 1; st >>= 1) s += __shfl_xor(s, st, 32);
  if (lane == 0) atomicAdd(&out[b], sqrtf(s) * (1.f / 384.f));
}

// e_xy: 8*1024 rows of ||Xf - Yf||
__global__ void k_exy(const float* __restrict__ X, const float* __restrict__ Y,
                      float* __restrict__ out) {
  const int wave = blockIdx.x * (blockDim.x >> 5) + (threadIdx.x >> 5);
  const int lane = threadIdx.x & 31;
  const int b = wave >> 10, n = wave & 1023;
  const int c = n >> 4, p = n & 15, rx = p >> 2, ry = p & 3;
  const float* x = X + ((b * C_ + c) * R_ + rx) * D_;
  const float* y = Y + ((b * C_ + c) * R_ + ry) * D_;
  float s = 0.f;
  for (int t = lane; t < D_; t += 32) { float d = x[t] - y[t]; s += d * d; }
  #pragma unroll
  for (int st = 16; st >= 1; st >>= 1) s += __shfl_xor(s, st, 32);
  if (lane == 0) atomicAdd(&out[b], sqrtf(s) * (1.f / 1024.f));
}

__global__ void k_loss(const float* __restrict__ eacc, float* __restrict__ out) {
  if (threadIdx.x == 0 && blockIdx.x == 0) {
    float l = 0.f;
    for (int b = 0; b < B_; ++b) {
      const float v = eacc[16 + b] - 0.5f * (eacc[b] + eacc[8 + b]);
      l += sqrtf(fmaxf(v, 0.f));
    }
    out[0] = l * (1.f / (float)B_);
  }
}

__global__ void k_copyT(const float* __restrict__ T, float* __restrict__ out) {
  const int i = blockIdx.x * blockDim.x + threadIdx.x;
  out[1 + i] = T[i];
}

// ---------------------------------------------------------------------------
extern "C" void kernel_launch(void* const* d_in, const int* in_sizes, int n_in,
                              void* d_out, int out_size, void* d_ws, size_t ws_size,
                              hipStream_t stream) {
  const float* X = (const float*)d_in[0];
  const float* Y = (const float*)d_in[1];
  float* out = (float*)d_out;

  float* ws    = (float*)d_ws;
  float* bufT  = ws;                          // 8*256*256  (T ping)
  float* bufA  = bufT + B_ * TD2_;            // 8*256*256  (M, then T^T T)
  float* bufN  = bufA + B_ * TD2_;            // 8*256*256  (T pong)
  float* w2    = bufN + B_ * TD2_;            // 8*1024
  float* rsum  = w2 + B_ * NF_;               // 8*1024
  float* eacc  = rsum + B_ * NF_;             // exx[8], eyy[8], exy[8]

  // ---- one-time: weights = 1, zeroed accumulators, loop-invariant energies -
  k_fill<<<(B_ * NF_ + 255) / 256, 256, 0, stream>>>(w2, 1.f, B_ * NF_);
  k_fill<<<(B_ * NF_ + 255) / 256, 256, 0, stream>>>(rsum, 0.f, B_ * NF_);
  k_fill<<<1, 32, 0, stream>>>(eacc, 0.f, 24);
  k_pair_energy<<<384, 256, 0, stream>>>(X, eacc);        // e_xx
  k_pair_energy<<<384, 256, 0, stream>>>(Y, eacc + 8);    // e_yy
  k_exy<<<1024, 256, 0, stream>>>(X, Y, eacc + 16);       // e_xy

  // ---- 100 iterations, 4 launches each --------------------------------------
  for (int it = 0; it < NITER; ++it) {
    k_gemm_M<<<256, 256, 0, stream>>>(X, Y, w2, bufA);          // M -> bufA
    k_ns_polar<<<B_, 1024, 0, stream>>>(bufA, bufT, bufN, bufA); // T -> bufT
    k_resid<<<1024, 256, 0, stream>>>(X, Y, bufT, rsum);
    k_w2up<<<(B_ * NF_ + 255) / 256, 256, 0, stream>>>(rsum, w2);
  }

  // ---- outputs: [loss, T(8x256x256)] ---------------------------------------
  k_loss<<<1, 32, 0, stream>>>(eacc, out);
  k_copyT<<<(B_ * TD2_) / 256, 256, 0, stream>>>(bufT, out);
  (void)in_sizes; (void)n_in; (void)out_size; (void)ws_size;
}